// GyroLoss_61280593379802
// MI455X (gfx1250) — compile-verified
//
#include <hip/hip_runtime.h>
#include <hip/hip_bf16.h>
#include <math.h>

typedef __attribute__((ext_vector_type(2))) float v2f;
typedef __attribute__((ext_vector_type(8))) float v8f;

#define DT_      0.005f
#define HUBER_   0.005f
#define BETA_    0.005f
// level-4: 32 * (2048-5) * 3 elements ; level-5: 32 * (1024-5) * 3 elements
#define CNT4_    196128.0f
#define CNT5_    97824.0f

// ---------------- small SO(3) helpers (row-major 3x3 in float[9]) ----------------

__device__ __forceinline__ void so3_exp9(float x, float y, float z, float* R) {
    float th2  = x * x + y * y + z * z;
    float th   = sqrtf(th2 + 1e-24f);
    float a, b;
    if (th2 < 1e-12f) {
        a = 1.0f - th2 * (1.0f / 6.0f);
        b = 0.5f - th2 * (1.0f / 24.0f);
    } else {
        a = sinf(th) / th;
        b = (1.0f - cosf(th)) / th2;
    }
    float xx = x * x, yy = y * y, zz = z * z;
    float xy = x * y, xz = x * z, yz = y * z;
    R[0] = 1.0f - b * (yy + zz);
    R[1] = -a * z + b * xy;
    R[2] =  a * y + b * xz;
    R[3] =  a * z + b * xy;
    R[4] = 1.0f - b * (xx + zz);
    R[5] = -a * x + b * yz;
    R[6] = -a * y + b * xz;
    R[7] =  a * x + b * yz;
    R[8] = 1.0f - b * (xx + yy);
}

__device__ __forceinline__ void mm3(const float* A, const float* B, float* C) {
#pragma unroll
    for (int i = 0; i < 3; ++i) {
        float a0 = A[3 * i], a1 = A[3 * i + 1], a2 = A[3 * i + 2];
        C[3 * i + 0] = fmaf(a0, B[0], fmaf(a1, B[3], a2 * B[6]));
        C[3 * i + 1] = fmaf(a0, B[1], fmaf(a1, B[4], a2 * B[7]));
        C[3 * i + 2] = fmaf(a0, B[2], fmaf(a1, B[5], a2 * B[8]));
    }
}

// C = A^T * B   (bmtm)
__device__ __forceinline__ void mtm3(const float* A, const float* B, float* C) {
#pragma unroll
    for (int i = 0; i < 3; ++i) {
        float a0 = A[i], a1 = A[3 + i], a2 = A[6 + i];
        C[3 * i + 0] = fmaf(a0, B[0], fmaf(a1, B[3], a2 * B[6]));
        C[3 * i + 1] = fmaf(a0, B[1], fmaf(a1, B[4], a2 * B[7]));
        C[3 * i + 2] = fmaf(a0, B[2], fmaf(a1, B[5], a2 * B[8]));
    }
}

__device__ __forceinline__ void so3_log9(const float* R, float* w) {
    float tr = R[0] + R[4] + R[8];
    float c  = 0.5f * (tr - 1.0f);
    c = fminf(fmaxf(c, -1.0f + 1e-7f), 1.0f - 1e-7f);
    float th = acosf(c);
    float sc = th / sinf(th);
    w[0] = 0.5f * sc * (R[7] - R[5]);
    w[1] = 0.5f * sc * (R[2] - R[6]);
    w[2] = 0.5f * sc * (R[3] - R[1]);
}

__device__ __forceinline__ float huber3(const float* w) {
    float s = 0.0f;
#pragma unroll
    for (int c = 0; c < 3; ++c) {
        float ax = fabsf(w[c]) * (1.0f / HUBER_);
        s += (ax < BETA_) ? (0.5f * ax * ax * (1.0f / BETA_)) : (ax - 0.5f * BETA_);
    }
    return s;
}

// ------------- exact f32 wave32 sum via V_WMMA_F32_16X16X4_F32 -------------
// A gets (v, 0) per lane -> the 64 A-slots hold all 32 lane values plus zeros.
// B = ones (layout independent). D[m,n] = rowsum_m(A), so any full column of D
// sums to the total. Lanes l and l^16 hold one full column of C/D between them.
__device__ __forceinline__ float wave_sum32(float v) {
    v2f a; a[0] = v;    a[1] = 0.0f;
    v2f b; b[0] = 1.0f; b[1] = 1.0f;
    v8f c = {};
    c = __builtin_amdgcn_wmma_f32_16x16x4_f32(
            /*neg_a=*/false, a, /*neg_b=*/false, b,
            /*c_mod=*/(short)0, c, /*reuse_a=*/false, /*reuse_b=*/false);
    float s = ((c[0] + c[1]) + (c[2] + c[3])) + ((c[4] + c[5]) + (c[6] + c[7]));
    s += __shfl_xor(s, 16, 32);
    return s;   // every lane holds the exact 32-lane total
}

// ---------------------------- kernels ----------------------------

__global__ __launch_bounds__(256) void gyro_main(const float* __restrict__ xs,
                                                 const float* __restrict__ hat,
                                                 float* __restrict__ partial) {
    const int g = blockIdx.x * 256 + threadIdx.x;   // 0..65535, group = (n, i)
    const int i = g & 2047;                         // index within sequence (level 4)

    // 16 consecutive gyro samples: 48 contiguous floats, 16B aligned (192*g bytes)
    const float4* hp = (const float4*)hat + 12 * (size_t)g;
    float h[48];
#pragma unroll
    for (int q = 0; q < 12; ++q) {
        float4 v = hp[q];
        h[4 * q + 0] = v.x; h[4 * q + 1] = v.y; h[4 * q + 2] = v.z; h[4 * q + 3] = v.w;
    }

    // Om16 = exp(DT*h0) * exp(DT*h1) * ... * exp(DT*h15)
    float Om[9], E[9], T[9];
    so3_exp9(DT_ * h[0], DT_ * h[1], DT_ * h[2], Om);
#pragma unroll
    for (int j = 1; j < 16; ++j) {
        so3_exp9(DT_ * h[3 * j], DT_ * h[3 * j + 1], DT_ * h[3 * j + 2], E);
        mm3(Om, E, T);
#pragma unroll
        for (int q = 0; q < 9; ++q) Om[q] = T[q];
    }

    // Xs16 = exp(xs[n, 16*i])  (floats 48g..48g+2, float4-aligned load)
    float4 xv = *((const float4*)xs + 12 * (size_t)g);
    float X[9];
    so3_exp9(xv.x, xv.y, xv.z, X);

    // ---- level 4 residual ----
    float R4[9], w4[3];
    mtm3(Om, X, R4);
    so3_log9(R4, w4);
    float sum4 = (i >= 5) ? huber3(w4) : 0.0f;

    // ---- level 5: pair adjacent lanes (even lane owns the pair) ----
    float Po[9], Px[9];
#pragma unroll
    for (int q = 0; q < 9; ++q) {
        Po[q] = __shfl_xor(Om[q], 1, 32);
        Px[q] = __shfl_xor(X[q], 1, 32);
    }
    float Om32[9], X32[9], R5[9], w5[3];
    mm3(Om, Po, Om32);   // correct ordering on even lanes; odd lanes masked below
    mm3(X,  Px, X32);
    mtm3(Om32, X32, R5);
    so3_log9(R5, w5);
    float sum5 = (((i & 1) == 0) && (i >= 10)) ? huber3(w5) : 0.0f;

    // ---- exact wave reduction through the matrix core (EXEC all-1s here) ----
    float S4 = wave_sum32(sum4);
    float S5 = wave_sum32(sum5);

    __shared__ float sm[16];
    const int lane = threadIdx.x & 31, wid = threadIdx.x >> 5;
    if (lane == 0) { sm[2 * wid] = S4; sm[2 * wid + 1] = S5; }
    __syncthreads();
    if (threadIdx.x == 0) {
        float t4 = 0.0f, t5 = 0.0f;
#pragma unroll
        for (int q = 0; q < 8; ++q) { t4 += sm[2 * q]; t5 += sm[2 * q + 1]; }
        partial[2 * blockIdx.x]     = t4;
        partial[2 * blockIdx.x + 1] = t5;
    }
}

__global__ __launch_bounds__(256) void gyro_final(const float* __restrict__ partial,
                                                  float* __restrict__ out) {
    float s4 = partial[2 * threadIdx.x];
    float s5 = partial[2 * threadIdx.x + 1];

    float S4 = wave_sum32(s4);
    float S5 = wave_sum32(s5);

    __shared__ float sm[16];
    const int lane = threadIdx.x & 31, wid = threadIdx.x >> 5;
    if (lane == 0) { sm[2 * wid] = S4; sm[2 * wid + 1] = S5; }
    __syncthreads();
    if (threadIdx.x == 0) {
        float t4 = 0.0f, t5 = 0.0f;
#pragma unroll
        for (int q = 0; q < 8; ++q) { t4 += sm[2 * q]; t5 += sm[2 * q + 1]; }
        // loss = W*HUBER^2 * mean4 + (W*HUBER^2/2) * mean5
        out[0] = 25.0f * (t4 / CNT4_) + 12.5f * (t5 / CNT5_);
    }
}

extern "C" void kernel_launch(void* const* d_in, const int* in_sizes, int n_in,
                              void* d_out, int out_size, void* d_ws, size_t ws_size,
                              hipStream_t stream) {
    const float* xs  = (const float*)d_in[0];
    const float* hat = (const float*)d_in[1];
    float* partial   = (float*)d_ws;        // 512 floats of scratch, fully overwritten
    gyro_main <<<256, 256, 0, stream>>>(xs, hat, partial);
    gyro_final<<<1,   256, 0, stream>>>(partial, (float*)d_out);
}